// FWIForward_68040871903321
// MI455X (gfx1250) — compile-verified
//
#include <hip/hip_runtime.h>
#include <cstdint>

// FWI forward modeling for MI455X (gfx1250).
// 10 blocks = (batch 0..1) x (shot 0..4); 768 threads (24 wave32) per block.
// Double-buffered wavefield in LDS (2 x 90KB); velocity staging aliases pB.
// One s_barrier per time step; ~65 VGPRs persistent -> no scratch spills.

#define NTSTEPS 1000

__global__ __launch_bounds__(768, 1)
void fwi_fwd_kernel(const float* __restrict__ v, float* __restrict__ out)
{
    __shared__ float pA[22500];    // wavefield buffer A (150 x 150)
    __shared__ float pB[22500];    // wavefield buffer B (150 x 150)
    __shared__ float srcw[1000];   // ricker wavelet
    __shared__ int   vminbits;

    const int tid = threadIdx.x;
    const int blk = blockIdx.x;        // 0..9 == b*5 + s
    const int b   = blk / 5;
    const int s   = blk - b * 5;
    const float* vb = v + b * 4900;

    float* const vld = pB;             // velocity staging aliases pB (setup only)

    // ---- stage velocity into LDS via gfx1250 async DMA path ----
    for (int i = tid; i < 4900; i += 768) {
        unsigned ldsAddr = (unsigned)(uintptr_t)(&vld[i]);   // LDS byte offset
        unsigned gOff    = (unsigned)(i * 4);
        asm volatile("global_load_async_to_lds_b32 %0, %1, %2"
                     :: "v"(ldsAddr), "v"(gOff), "s"(vb) : "memory");
    }

    // build ricker wavelet while the async copy is in flight
    for (int i = tid; i < 1000; i += 768) {
        float w = 0.0f;
        if (i < 147) {  // nw = 2*floor((2.2/f/dt)/2)+1 = 147, nc = 73
            double a  = (73.0 - (double)i) * (0.015 * 3.14159265358979323846);
            double bb = a * a;
            w = (float)((1.0 - 2.0 * bb) * exp(-bb));
        }
        srcw[i] = w;
    }
    if (tid == 0) vminbits = 0x7f7fffff;  // +FLT_MAX bits

    asm volatile("s_wait_asynccnt 0x0" ::: "memory");
    __syncthreads();

    // ---- velmin reduction (positive floats: int order == float order) ----
    float m = 3.402823466e38f;
    for (int i = tid; i < 4900; i += 768) m = fminf(m, vld[i]);
    atomicMin(&vminbits, __float_as_int(m));
    __syncthreads();
    const float velmin = __int_as_float(vminbits);
    const float kappa  = 3.0f * velmin * logf(1.0e7f) / (2.0f * 390.0f); // a=(NBC-1)*DX
    const float kf     = kappa * (10.0f / 390.0f) * (10.0f / 390.0f);    // kappa*(DX/a)^2

    // source columns: isx = around(linspace(0,69,5)) + 40 (bankers rounding)
    const int sx_tab[5] = {40, 57, 74, 92, 109};
    const int sxc = sx_tab[s];
    float bsv = vld[1 * 70 + (sxc - 40)] * 0.001f;  // v_pad(z=41,x=sxc) * DT
    const float bsrcv = bsv * bsv;

    // ---- thread decomposition: 150 rows x 5 segments of 30 cells ----
    const bool active = tid < 750;
    const int z   = tid / 5;
    const int x0  = (tid - z * 5) * 30;
    const int row = z * 150;
    const int rU1 = ((z == 0)   ? 149     : z - 1) * 150;  // jnp.roll wrap
    const int rU2 = ((z <= 1)   ? z + 148 : z - 2) * 150;
    const int rD1 = ((z == 149) ? 0       : z + 1) * 150;
    const int rD2 = ((z >= 148) ? z - 148 : z + 2) * 150;

    // z-border damping (used only where x is interior); x-borders override
    float zdamp = 0.0f;
    if      (z < 40)   { float c = (float)(39 - z);  zdamp = kf * c * c; }
    else if (z >= 110) { float c = (float)(z - 110); zdamp = kf * c * c; }

    // alpha = (v_pad*DT/DX)^2 : only persistent per-cell array (30 VGPRs)
    float al[30];
    if (active) {
#pragma unroll
        for (int j = 0; j < 30; ++j) {
            const int x = x0 + j;
            int zz = z - 40; zz = zz < 0 ? 0 : (zz > 69 ? 69 : zz);  // edge pad
            int xx = x - 40; xx = xx < 0 ? 0 : (xx > 69 ? 69 : xx);
            const float vp = vld[zz * 70 + xx];
            float a = vp * 1.0e-4f;
            al[j] = a * a;
        }
    }
    __syncthreads();   // all reads of vld done; pB may now be reused

    // zero both wavefield buffers (p0 = p1 = 0)
    for (int i = tid; i < 22500; i += 768) { pA[i] = 0.0f; pB[i] = 0.0f; }
    __syncthreads();

    const bool recRow = active && (z == 41);               // igz == isz == 41
    float* outR = out + (size_t)blk * (size_t)(NTSTEPS * 70);

    const float C2F = 4.0f / 3.0f, C3F = -1.0f / 12.0f;

    for (int t = 0; t < NTSTEPS; ++t) {
        float* const W = (t & 1) ? pB : pA;   // holds p0 (read own cell), gets new p
        float* const R = (t & 1) ? pA : pB;   // holds p1 (read neighborhood)
        const float sval = srcw[t];
        if (active) {
            // sliding x-window over own row (periodic wrap only at segment edges)
            float w0 = R[row + (x0 >= 2 ? x0 - 2 : x0 + 148)];
            float w1 = R[row + (x0 >= 1 ? x0 - 1 : x0 + 149)];
            float w2 = R[row + x0];
            float w3 = R[row + x0 + 1];
#pragma unroll
            for (int j = 0; j < 30; ++j) {
                const int x = x0 + j;
                int xp2 = x + 2; if (xp2 >= 150) xp2 -= 150;
                const float w4 = R[row + xp2];
                const float u1 = R[rU1 + x];
                const float d1 = R[rD1 + x];
                const float u2 = R[rU2 + x];
                const float d2 = R[rD2 + x];
                const float lap = C2F * (w1 + w3 + u1 + d1) + C3F * (w0 + w4 + u2 + d2);
                // damping: x-border overrides z-border (reference .set order)
                float d;
                if      (x < 40)   { float c = (float)(39 - x);  d = kf * c * c; }
                else if (x >= 110) { float c = (float)(x - 110); d = kf * c * c; }
                else               d = zdamp;
                const float kdt = d * 0.001f;                 // damp * DT
                const float t2v = 1.0f - kdt;
                const float t1v = 2.0f - kdt - 5.0f * al[j];  // 2 + 2*C1*alpha - kdt
                const float p0v = W[row + x];                 // own cell only
                float val = t1v * w2 - t2v * p0v + al[j] * lap;
                if (recRow && x == sxc) val += bsrcv * sval;  // inject before record
                W[row + x] = val;
                if (recRow && x >= 40 && x < 110)             // receivers 40..109
                    outR[t * 70 + (x - 40)] = val;
                w0 = w1; w1 = w2; w2 = w3; w3 = w4;
            }
        }
        __syncthreads();   // new p visible to all before next step's reads
    }
}

extern "C" void kernel_launch(void* const* d_in, const int* in_sizes, int n_in,
                              void* d_out, int out_size, void* d_ws, size_t ws_size,
                              hipStream_t stream)
{
    (void)in_sizes; (void)n_in; (void)d_ws; (void)ws_size; (void)out_size;
    const float* v   = (const float*)d_in[0];
    float*       out = (float*)d_out;
    fwi_fwd_kernel<<<dim3(10), dim3(768), 0, stream>>>(v, out);
}